// ModelVLLM_70471823393008
// MI455X (gfx1250) — compile-verified
//
#include <hip/hip_runtime.h>
#include <cstdint>
#include <cstddef>

// Block-wise recurrent KV state reduction (vLLM lightning-attn _fwd_kv_reduce)
//   for i in 0..NB-1:  state = exp(-s[h]*min(seq_len-i*64,64)) * state + kv[b,h,i]
// Shapes (from setup_inputs): s[32], kv[4,32,128,64,64] f32, kv_history[4,32,64,64] f32.
// Memory-bound (268 MB @ 23.3 TB/s ~ 11.5 us); implemented as an async-to-LDS
// streaming pipeline (CDNA5 GLOBAL_LOAD_ASYNC_TO_LDS_B128 + s_wait_asynccnt).

constexpr int BLOCK_T = 64;            // lightning-attn block size (matches reference)
constexpr int TILE    = 64 * 64;       // d*e state tile, in floats (reference: d=e=64)
constexpr int SPLIT   = 4;             // workgroups per (b,h) pair
constexpr int CHUNK   = TILE / SPLIT;  // 1024 floats = 4 KB per workgroup per block
constexpr int THREADS = 256;           // 8 waves (wave32)
constexpr int VPT     = CHUNK / THREADS;  // 4 floats (one b128) per thread
constexpr int DEPTH   = 8;             // async pipeline depth (32 KB LDS ring)
constexpr int DECMAX  = 256;           // max NB for precomputed decay table

// ---- CDNA5 async-copy primitives (gfx1250) ---------------------------------
__device__ __forceinline__ void async_load_b128(unsigned lds_byte_off,
                                                const float* gptr) {
    // GLOBAL_LOAD_ASYNC_TO_LDS_B128 (GV mode): vdst = per-lane LDS byte addr,
    // vaddr = per-lane 64-bit global addr. Tracked with ASYNCcnt.
    asm volatile("global_load_async_to_lds_b128 %0, %1, off"
                 :: "v"(lds_byte_off), "v"(gptr)
                 : "memory");
}

#define WAIT_ASYNC(N) asm volatile("s_wait_asynccnt %0" :: "n"(N) : "memory")
#define WAIT_DS0()    asm volatile("s_wait_dscnt 0x0" ::: "memory")

__device__ __forceinline__ unsigned lds_byte_addr(const void* p) {
    // Generic (flat) pointers into the LDS aperture carry the wave-relative
    // LDS byte offset in their low 32 bits (ISA 10.2 aperture mapping).
    return (unsigned)(uintptr_t)p;
}

// ----------------------------------------------------------------------------
__global__ __launch_bounds__(THREADS)
void kv_reduce_async_kernel(const float* __restrict__ s,
                            const float* __restrict__ kv,
                            const float* __restrict__ hist,
                            const int*   __restrict__ seq_len_p,
                            float* __restrict__ out,
                            int H, int NB) {
    __shared__ float lds_ring[DEPTH][CHUNK];  // 32 KB async staging ring
    __shared__ float s_dec[DECMAX];           // per-block decay factors

    const int wg   = blockIdx.x;
    const int pair = wg / SPLIT;              // flat (b*H + h)
    const int part = wg % SPLIT;              // which d-quarter of the tile
    const int h    = pair % H;
    const int tid  = threadIdx.x;

    const int   seq_len = *seq_len_p;
    const float sh      = s[h];

    // Precompute decays once (uniform per workgroup); one barrier total.
    for (int i = tid; i < NB && i < DECMAX; i += THREADS) {
        float bs = fminf((float)(seq_len - i * BLOCK_T), (float)BLOCK_T);
        s_dec[i] = expf(-sh * bs);
    }
    __syncthreads();

    // Per-thread addressing: thread owns 4 consecutive floats of its chunk.
    const size_t chunk_off = (size_t)part * CHUNK + (size_t)tid * VPT;
    const size_t tile_off  = (size_t)pair * TILE + chunk_off;
    const float* kv_base   = kv + (size_t)pair * (size_t)NB * TILE + chunk_off;

    // Seed state from kv_history (registers).
    float4 state = *(const float4*)(hist + tile_off);

    const unsigned lds0 = lds_byte_addr(&lds_ring[0][tid * VPT]);

    // Prologue: fill the async pipeline.
    const int prefill = (NB < DEPTH) ? NB : DEPTH;
    for (int p = 0; p < prefill; ++p) {
        async_load_b128(lds0 + (unsigned)(p * CHUNK * sizeof(float)),
                        kv_base + (size_t)p * TILE);
    }

    // Steady state: wait oldest, FMA, refill slot. Each thread reads only the
    // LDS bytes its own lane requested -> no intra-loop barriers needed.
    for (int i = 0; i < NB; ++i) {
        const int slot = i & (DEPTH - 1);

        if (i < NB - DEPTH) { WAIT_ASYNC(DEPTH - 1); }  // load i retired
        else                { WAIT_ASYNC(0); }          // drain tail

        const float4 kvv = *(const float4*)&lds_ring[slot][tid * VPT];
        const float  dec = (i < DECMAX)
                         ? s_dec[i]
                         : expf(-sh * fminf((float)(seq_len - i * BLOCK_T),
                                            (float)BLOCK_T));

        state.x = fmaf(dec, state.x, kvv.x);
        state.y = fmaf(dec, state.y, kvv.y);
        state.z = fmaf(dec, state.z, kvv.z);
        state.w = fmaf(dec, state.w, kvv.w);

        const int nxt = i + DEPTH;
        if (nxt < NB) {
            // Ensure our ds_load of this slot retired before the async engine
            // may overwrite it, then re-issue into the freed slot.
            WAIT_DS0();
            async_load_b128(lds0 + (unsigned)(slot * CHUNK * sizeof(float)),
                            kv_base + (size_t)nxt * TILE);
        }
    }

    *(float4*)(out + tile_off) = state;
}

// ----------------------------------------------------------------------------
extern "C" void kernel_launch(void* const* d_in, const int* in_sizes, int n_in,
                              void* d_out, int out_size, void* d_ws, size_t ws_size,
                              hipStream_t stream) {
    const float* s       = (const float*)d_in[0];   // [h]
    const float* kv      = (const float*)d_in[1];   // [b,h,nb,d,e]
    const float* hist    = (const float*)d_in[2];   // [b,h,d,e]
    const int*   seq_len = (const int*)d_in[3];     // scalar on device

    const int H  = in_sizes[0];
    const int BH = in_sizes[2] / TILE;              // b*h pairs
    const int NB = in_sizes[1] / in_sizes[2];       // num_block

    float* out = (float*)d_out;

    dim3 grid(BH * SPLIT);
    dim3 block(THREADS);
    kv_reduce_async_kernel<<<grid, block, 0, stream>>>(s, kv, hist, seq_len,
                                                       out, H, NB);
}